// KuramotoSivashinsky_7705171329353
// MI455X (gfx1250) — compile-verified
//
#include <hip/hip_runtime.h>

// Kuramoto-Sivashinsky ETDRK2 spectral integrator, NX=1024, nt=8192.
// Single persistent workgroup (256 threads = 8 wave32s) on one WGP; all state
// resident in LDS. 1024-pt FFT = four-step (32x32): two 32x32 complex matmuls
// on the f32 WMMA pipe (v_wmma_f32_16x16x4_f32) + a register-resident twiddle
// pass. DFT-32 matrix fragments hoisted into VGPRs (loop-invariant); FFT has
// separate IN/OUT planes so the step needs zero state-copy loops (3-way plane
// pointer rotation instead).

typedef __attribute__((ext_vector_type(2))) float v2f;
typedef __attribute__((ext_vector_type(8))) float v8f;

#define ROWS  32
#define PITCH 33          // pad to kill LDS bank conflicts on column accesses
#define PLANE (ROWS * PITCH)
#define NTHREADS 256

__device__ __forceinline__ v8f wmma4(v2f a, v2f b, v8f c) {
    return __builtin_amdgcn_wmma_f32_16x16x4_f32(false, a, false, b,
                                                 (short)0, c, false, false);
}
__device__ __forceinline__ v2f negv(v2f v) { v2f r; r.x = -v.x; r.y = -v.y; return r; }

// ---- WMMA fragment loaders (V_WMMA_F32_16X16X4_F32 layouts, ISA 7.12.2) ----
// A 16x4: lanes 0-15 -> M=lane, VGPR0=K0,VGPR1=K1 ; lanes 16-31 -> K2,K3
__device__ __forceinline__ v2f load_a_frag(const float* A, int mBase, int k0, int lane) {
    int m  = mBase + (lane & 15);
    int kb = k0 + ((lane >> 4) << 1);
    v2f a;
    a.x = A[m * PITCH + kb + 0];
    a.y = A[m * PITCH + kb + 1];
    return a;
}
// B 4x16: rows striped across lanes within a VGPR (C-layout analogue)
__device__ __forceinline__ v2f load_b_frag(const float* B, int nBase, int k0, int lane) {
    int n  = nBase + (lane & 15);
    int kb = k0 + ((lane >> 4) << 1);
    v2f b;
    b.x = B[(kb + 0) * PITCH + n];
    b.y = B[(kb + 1) * PITCH + n];
    return b;
}

// Hoisted DFT-32 matrix fragments: per wave, only its own tile bases needed.
struct WFrags {
    v2f bre[8], bim[8];   // B-side fragments (stage A: C = data x W), at nBase
    v2f are[8], aim[8];   // A-side fragments (stage B: C = W x data), at mBase
};

// ---- Stage A: C = data x (Wre + CONJ*i*Wim); data fragments from LDS -------
template<int CONJ>
__device__ __forceinline__ void cmatmul_dataW(const float* Are, const float* Aim,
                                              const WFrags& wf,
                                              float* Cre, float* Cim,
                                              int lane, int wave) {
    const int mBase = ((wave >> 2) & 1) << 4;
    const int nBase = ((wave >> 1) & 1) << 4;
    const int part  = wave & 1;
    v8f acc = {};
    if (part == 0) {
        // Cre = Ar*Wr - CONJ * Ai*Wi   (sign folded into the Ai fragment)
        #pragma unroll
        for (int k = 0; k < 8; ++k) {
            v2f a = load_a_frag(Are, mBase, 4 * k, lane);
            acc = wmma4(a, wf.bre[k], acc);
            v2f ai = load_a_frag(Aim, mBase, 4 * k, lane);
            if (CONJ > 0) ai = negv(ai);
            acc = wmma4(ai, wf.bim[k], acc);
        }
    } else {
        // Cim = CONJ * Ar*Wi + Ai*Wr   (sign folded into the register Wi frag)
        #pragma unroll
        for (int k = 0; k < 8; ++k) {
            v2f a  = load_a_frag(Are, mBase, 4 * k, lane);
            v2f bw = (CONJ > 0) ? wf.bim[k] : negv(wf.bim[k]);
            acc = wmma4(a, bw, acc);
            v2f ai = load_a_frag(Aim, mBase, 4 * k, lane);
            acc = wmma4(ai, wf.bre[k], acc);
        }
    }
    float* C = part ? Cim : Cre;
    const int col  = nBase + (lane & 15);
    const int rOff = (lane >> 4) << 3;
    #pragma unroll
    for (int r = 0; r < 8; ++r)
        C[(mBase + rOff + r) * PITCH + col] = acc[r];
}

// ---- Stage B: C = (Wre + CONJ*i*Wim) x data; data fragments from LDS -------
template<int CONJ>
__device__ __forceinline__ void cmatmul_Wdata(const WFrags& wf,
                                              const float* Bre, const float* Bim,
                                              float* Cre, float* Cim,
                                              float scale, int lane, int wave) {
    const int mBase = ((wave >> 2) & 1) << 4;
    const int nBase = ((wave >> 1) & 1) << 4;
    const int part  = wave & 1;
    v8f acc = {};
    if (part == 0) {
        // Cre = Wr*Br - CONJ * Wi*Bi
        #pragma unroll
        for (int k = 0; k < 8; ++k) {
            v2f b = load_b_frag(Bre, nBase, 4 * k, lane);
            acc = wmma4(wf.are[k], b, acc);
            v2f aw = (CONJ > 0) ? negv(wf.aim[k]) : wf.aim[k];
            v2f bi = load_b_frag(Bim, nBase, 4 * k, lane);
            acc = wmma4(aw, bi, acc);
        }
    } else {
        // Cim = Wr*Bi + CONJ * Wi*Br
        #pragma unroll
        for (int k = 0; k < 8; ++k) {
            v2f bi = load_b_frag(Bim, nBase, 4 * k, lane);
            acc = wmma4(wf.are[k], bi, acc);
            v2f aw = (CONJ > 0) ? wf.aim[k] : negv(wf.aim[k]);
            v2f b  = load_b_frag(Bre, nBase, 4 * k, lane);
            acc = wmma4(aw, b, acc);
        }
    }
    float* C = part ? Cim : Cre;
    const int col  = nBase + (lane & 15);
    const int rOff = (lane >> 4) << 3;
    #pragma unroll
    for (int r = 0; r < 8; ++r)
        C[(mBase + rOff + r) * PITCH + col] = acc[r] * scale;
}

// ---- 1024-pt FFT via four-step: OUT <- DFT(IN). CONJ=+1 fwd, -1 inv -------
// Stage A only READS IN (state-preserving); stage B only WRITES OUT.
// IN==OUT is legal (stage A drains IN into S1 before stage B overwrites OUT).
template<int CONJ>
__device__ __forceinline__ void fft1024(const float* INre, const float* INim,
                                        float* OUTre, float* OUTim,
                                        float* S1re, float* S1im,
                                        const WFrags& wf,
                                        const float* twr, const float* twi,
                                        float scale, int tid, int lane, int wave) {
    cmatmul_dataW<CONJ>(INre, INim, wf, S1re, S1im, lane, wave);
    __syncthreads();
    // Twiddle: S1[n1][k2] *= W1024^(+-n1*k2)  (factors live in registers)
    #pragma unroll
    for (int q = 0; q < 4; ++q) {
        int j   = tid + q * NTHREADS;
        int idx = (j >> 5) * PITCH + (j & 31);
        float tr = twr[q];
        float ti = (CONJ > 0) ? twi[q] : -twi[q];
        float xr = S1re[idx], xi = S1im[idx];
        S1re[idx] = xr * tr - xi * ti;
        S1im[idx] = xr * ti + xi * tr;
    }
    __syncthreads();
    cmatmul_Wdata<CONJ>(wf, S1re, S1im, OUTre, OUTim, scale, lane, wave);
    __syncthreads();
}

__global__ void __launch_bounds__(NTHREADS)
ks_etdrk2_kernel(const int* __restrict__ nt_ptr,
                 const float* __restrict__ u0,
                 float* __restrict__ out) {
    __shared__ float Wre[PLANE],  Wim[PLANE];     // DFT-32 matrix (init only)
    __shared__ float UHre[PLANE], UHim[PLANE];    // uhat state (never destroyed)
    __shared__ float PAre[PLANE], PAim[PLANE];    // rotating: uh2cur/uh2last/work
    __shared__ float PBre[PLANE], PBim[PLANE];
    __shared__ float PCre[PLANE], PCim[PLANE];
    __shared__ float S1re[PLANE], S1im[PLANE];    // FFT mid-stage scratch

    const int   nt   = *nt_ptr;
    const int   tid  = threadIdx.x;
    const int   lane = tid & 31;
    const int   wave = tid >> 5;
    const float TWO_PI = 6.283185307179586f;
    const float DTc = 0.05f, Lc = 100.0f;
    const float kScale = TWO_PI / Lc;
    const float invN   = 1.0f / 1024.0f;

    // ---- DFT-32 matrix into LDS: W[m][n] = exp(-2pi i mn/32) ----
    #pragma unroll
    for (int q = 0; q < 4; ++q) {
        int j   = tid + q * NTHREADS;
        int row = j >> 5, col = j & 31;
        int idx = row * PITCH + col;
        float aW = -TWO_PI * (float)((row * col) & 31) * (1.0f / 32.0f);
        Wre[idx] = cosf(aW);
        Wim[idx] = sinf(aW);
    }
    __syncthreads();

    // ---- hoist loop-invariant W fragments into VGPRs (per-wave tile bases) ----
    WFrags wf;
    {
        const int mBase = ((wave >> 2) & 1) << 4;
        const int nBase = ((wave >> 1) & 1) << 4;
        #pragma unroll
        for (int k = 0; k < 8; ++k) {
            wf.bre[k] = load_b_frag(Wre, nBase, 4 * k, lane);
            wf.bim[k] = load_b_frag(Wim, nBase, 4 * k, lane);
            wf.are[k] = load_a_frag(Wre, mBase, 4 * k, lane);
            wf.aim[k] = load_a_frag(Wim, mBase, 4 * k, lane);
        }
    }

    // ---- per-thread register constants: twiddles + ETDRK2 coefficients ----
    float twr[4], twi[4], c1[4], c2[4];
    #pragma unroll
    for (int q = 0; q < 4; ++q) {
        int j   = tid + q * NTHREADS;
        int row = j >> 5, col = j & 31;
        float aT = -TWO_PI * (float)((row * col) & 1023) * (1.0f / 1024.0f);
        twr[q] = cosf(aT);
        twi[q] = sinf(aT);
        // unshifted frequency (fftshift/ifftshift in the reference cancel)
        float kf  = (float)((j < 512) ? j : j - 1024);
        float w   = kScale * kf;
        float w2  = w * w;
        float FLv = w2 - w2 * w2;                 // NU = 1
        float invd = 1.0f / (1.0f - 0.5f * DTc * FLv);
        c1[q] = invd * (1.0f + 0.5f * DTc * FLv);
        c2[q] = invd * DTc * (-0.5f * kScale * kf);   // FN = i*fnc, c2=invd*DT*fnc
    }

    // ---- init: col 0 = u0; uhat = DFT(u0)/N; uhat2 = uhat2_last = DFT(u0^2)/N
    #pragma unroll
    for (int q = 0; q < 4; ++q) {
        int j   = tid + q * NTHREADS;
        int idx = (j >> 5) * PITCH + (j & 31);
        float v = u0[j];
        PAre[idx] = v; PAim[idx] = 0.0f;
        out[(size_t)j * (size_t)nt] = v;
    }
    __syncthreads();
    fft1024<1>(PAre, PAim, UHre, UHim, S1re, S1im, wf, twr, twi, invN,
               tid, lane, wave);
    #pragma unroll
    for (int q = 0; q < 4; ++q) {
        int j   = tid + q * NTHREADS;
        int idx = (j >> 5) * PITCH + (j & 31);
        float v = u0[j];
        PAre[idx] = v * v; PAim[idx] = 0.0f;
    }
    __syncthreads();
    fft1024<1>(PAre, PAim, PAre, PAim, S1re, S1im, wf, twr, twi, invN,
               tid, lane, wave);
    // un-alias uh2_last (init-only copy so the 3-way rotation never collides)
    #pragma unroll
    for (int q = 0; q < 4; ++q) {
        int j   = tid + q * NTHREADS;
        int idx = (j >> 5) * PITCH + (j & 31);
        PBre[idx] = PAre[idx]; PBim[idx] = PAim[idx];
    }
    __syncthreads();

    float *uh2re = PAre, *uh2im = PAim;     // current uhat2
    float *lre   = PBre, *lim   = PBim;     // uhat2_last
    float *wkre  = PCre, *wkim  = PCim;     // FFT work planes

    // ---- 8191 sequential ETDRK2 steps, state resident in LDS ----
    for (int t = 1; t < nt; ++t) {
        // uhat = c1*uhat + c2 * i-rotated (1.5*uhat2 - 0.5*uhat2_last), in place
        #pragma unroll
        for (int q = 0; q < 4; ++q) {
            int j   = tid + q * NTHREADS;
            int idx = (j >> 5) * PITCH + (j & 31);
            float gr = 1.5f * uh2re[idx] - 0.5f * lre[idx];
            float gi = 1.5f * uh2im[idx] - 0.5f * lim[idx];
            float ur = c1[q] * UHre[idx] - c2[q] * gi;
            float ui = c1[q] * UHim[idx] + c2[q] * gr;
            UHre[idx] = ur; UHim[idx] = ui;
        }
        __syncthreads();

        // u_next = real( N * ifft(uhat) ): conjugate DFT, UH preserved
        fft1024<-1>(UHre, UHim, wkre, wkim, S1re, S1im, wf, twr, twi, 1.0f,
                    tid, lane, wave);

        // stream output column t; square in place for the next forward FFT
        #pragma unroll
        for (int q = 0; q < 4; ++q) {
            int j   = tid + q * NTHREADS;
            int idx = (j >> 5) * PITCH + (j & 31);
            float v = wkre[idx];
            out[(size_t)j * (size_t)nt + (size_t)t] = v;
            wkre[idx] = v * v;
            wkim[idx] = 0.0f;
        }
        __syncthreads();

        // uhat2_next = DFT(u_next^2)/N, in place in the work planes
        fft1024<1>(wkre, wkim, wkre, wkim, S1re, S1im, wf, twr, twi, invN,
                   tid, lane, wave);

        // 3-way pointer rotation: cur <- work, last <- cur, work <- last
        float* t0 = lre;  float* t1 = lim;   // old last becomes new work
        lre = uh2re;  lim = uh2im;
        uh2re = wkre; uh2im = wkim;
        wkre = t0;    wkim = t1;
    }
}

extern "C" void kernel_launch(void* const* d_in, const int* in_sizes, int n_in,
                              void* d_out, int out_size, void* d_ws, size_t ws_size,
                              hipStream_t stream) {
    const int*   nt  = (const int*)d_in[0];
    const float* u0  = (const float*)d_in[1];
    float*       out = (float*)d_out;
    (void)in_sizes; (void)n_in; (void)out_size; (void)d_ws; (void)ws_size;
    hipLaunchKernelGGL(ks_etdrk2_kernel, dim3(1), dim3(NTHREADS), 0, stream,
                       nt, u0, out);
}